// SparseKVGather_50268297232601
// MI455X (gfx1250) — compile-verified
//
#include <hip/hip_runtime.h>
#include <hip/hip_bf16.h>
#include <stdint.h>

// Problem shape (fixed by the reference): B=1, H=32, S=8192, D=128, ACT=2048
#define HN   32
#define SN   8192
#define DN   128
#define ACTN 2048
#define CHUNKS_PER_ROW 32  // 128 floats * 4B / 16B per chunk
#define ITERS 4            // rows per wave, pipelined through the async DMA

// ---- feature detection: gfx1250 async global<->LDS DMA builtins -------------
#if defined(__has_builtin)
#if __has_builtin(__builtin_amdgcn_global_load_async_to_lds_b128) && \
    __has_builtin(__builtin_amdgcn_global_store_async_from_lds_b128)
#define USE_ASYNC_LDS 1
#endif
#endif

// Pointer types matching the builtin signature: v4i in as1 (global) / as3 (LDS)
typedef int v4i __attribute__((ext_vector_type(4)));
typedef __attribute__((address_space(1))) v4i* g_v4i_p;
typedef __attribute__((address_space(3))) v4i* l_v4i_p;

static __device__ __forceinline__ void wait_asynccnt0() {
#if defined(__has_builtin) && __has_builtin(__builtin_amdgcn_s_wait_asynccnt)
    __builtin_amdgcn_s_wait_asynccnt(0);
#else
    asm volatile("s_wait_asynccnt 0" ::: "memory");
#endif
}

// =============================================================================
// Kernel A: per-head stable compaction of activated indices.
// One block (1024 threads) per head; each thread owns 8 consecutive positions.
// Produces idx_out[h][r] = source position s for rank r, or -1 (zero-fill slot).
// =============================================================================
__global__ __launch_bounds__(1024)
void compact_indices_kernel(const int* __restrict__ act, int* __restrict__ idx_out) {
    const int h = blockIdx.x;
    const int t = threadIdx.x;

    __shared__ unsigned int sums[1024];

    const int* arow = act + (size_t)h * SN + (size_t)t * 8;
    unsigned int flags = 0;
#pragma unroll
    for (int j = 0; j < 8; ++j)
        flags |= (arow[j] != 0 ? 1u : 0u) << j;
    const unsigned int cnt = __popc(flags);

    sums[t] = cnt;
    __syncthreads();
    // Hillis-Steele inclusive scan over 1024 thread sums
    for (int off = 1; off < 1024; off <<= 1) {
        unsigned int vprev = (t >= off) ? sums[t - off] : 0u;
        __syncthreads();
        sums[t] += vprev;
        __syncthreads();
    }
    unsigned int rank = sums[t] - cnt;  // exclusive prefix

    // zero-fill slots first (slots beyond the activated count stay -1)
    int* row = idx_out + (size_t)h * ACTN;
    for (int i = t; i < ACTN; i += 1024) row[i] = -1;
    __syncthreads();

    const int s0 = t * 8;
#pragma unroll
    for (int j = 0; j < 8; ++j) {
        if ((flags >> j) & 1u) {
            if (rank < ACTN) row[rank] = s0 + j;
            ++rank;
        }
    }
}

// =============================================================================
// Kernel B: bandwidth kernel. One thread per 16-byte chunk; one wave (32 lanes)
// covers exactly one 512-byte K/V row, so the gathered index and the validity
// branch are wave-uniform (scalarized via readfirstlane) and every transaction
// is a coalesced B128.
// Valid rows ride the CDNA5 async DMA path (ASYNCcnt): 4 async loads into LDS,
// one wave-level s_wait_asynccnt, 4 async stores — 2 KB in flight per wave,
// never staging through VGPRs.
// =============================================================================
__global__ __launch_bounds__(256)
void gather_rows_kernel(const float* __restrict__ kin,
                        const float* __restrict__ vin,
                        const int* __restrict__ idx,
                        float* __restrict__ out) {
    const int t = threadIdx.x;
    const size_t nthreads = (size_t)gridDim.x * 256;   // = total_chunks / ITERS

#ifdef USE_ASYNC_LDS
    __shared__ float4 stage[256 * ITERS];
    float* dsts[ITERS];
#endif

    size_t g = (size_t)blockIdx.x * 256 + t;
#pragma unroll
    for (int i = 0; i < ITERS; ++i, g += nthreads) {
        const int    c     = (int)(g & (CHUNKS_PER_ROW - 1));
        const size_t row   = g >> 5;                           // / CHUNKS_PER_ROW
        const int    which = (int)(row / ((size_t)HN * ACTN)); // 0 = K, 1 = V
        const size_t rem   = row % ((size_t)HN * ACTN);
        const int    h     = (int)(rem / ACTN);
        const int    r     = (int)(rem % ACTN);

        // Wave-uniform by construction (32 chunks per row == 32 lanes per wave).
        const int src_s = __builtin_amdgcn_readfirstlane(idx[(size_t)h * ACTN + r]);

        const float* srcbase = which ? vin : kin;
        float* dst = out + (size_t)which * ((size_t)HN * ACTN * DN)
                         + ((size_t)h * ACTN + r) * DN + (size_t)c * 4;

#ifdef USE_ASYNC_LDS
        dsts[i] = dst;
        if (src_s >= 0) {
            const float* src = srcbase + ((size_t)h * SN + src_s) * DN + (size_t)c * 4;
            __builtin_amdgcn_global_load_async_to_lds_b128(
                (g_v4i_p)src, (l_v4i_p)&stage[t + 256 * i], 0, 0);
        } else {
            stage[t + 256 * i] = make_float4(0.f, 0.f, 0.f, 0.f); // ds_store_b128
        }
#else
        float4 val = make_float4(0.f, 0.f, 0.f, 0.f);
        if (src_s >= 0) {
            const float* src = srcbase + ((size_t)h * SN + src_s) * DN + (size_t)c * 4;
            val = *(const float4*)src;
        }
        *(float4*)dst = val;
#endif
    }

#ifdef USE_ASYNC_LDS
    // Drain all async loads (ASYNCcnt) and any LDS zero-stores (DScnt) before
    // the async store engine reads these LDS slots.
    wait_asynccnt0();
    asm volatile("s_wait_dscnt 0" ::: "memory");

#pragma unroll
    for (int i = 0; i < ITERS; ++i) {
        __builtin_amdgcn_global_store_async_from_lds_b128(
            (g_v4i_p)dsts[i], (l_v4i_p)&stage[t + 256 * i], 0, 0);
    }
    // S_ENDPGM performs an implicit wait-idle, draining the outstanding stores.
#endif
}

// =============================================================================
extern "C" void kernel_launch(void* const* d_in, const int* in_sizes, int n_in,
                              void* d_out, int out_size, void* d_ws, size_t ws_size,
                              hipStream_t stream) {
    const float* kin = (const float*)d_in[0];
    const float* vin = (const float*)d_in[1];
    const int*   act = (const int*)d_in[2];   // bool -> int32 0/1 per harness contract
    // d_in[3] = max_activated scalar (always 2048 for this shape)

    int* idx = (int*)d_ws;                    // 32*2048*4 = 256 KB scratch

    compact_indices_kernel<<<HN, 1024, 0, stream>>>(act, idx);

    const size_t total_chunks = (size_t)2 * HN * ACTN * CHUNKS_PER_ROW;  // 4,194,304
    const int    blocks       = (int)(total_chunks / (256 * ITERS));     // 4096
    gather_rows_kernel<<<blocks, 256, 0, stream>>>(kin, vin, idx, (float*)d_out);
}